// ReconstructorHGNN_85117661872490
// MI455X (gfx1250) — compile-verified
//
#include <hip/hip_runtime.h>

// Problem constants (match reference)
#define C_REL   5
#define N_NODES 50000
#define E_EDGES 800000
#define DIN_F   128
#define H_F     128
#define DOUT_F  64

typedef __attribute__((ext_vector_type(2))) float v2f;
typedef __attribute__((ext_vector_type(8))) float v8f;

__device__ __forceinline__ void atomic_add_f32(float* p, float v) {
    __hip_atomic_fetch_add(p, v, __ATOMIC_RELAXED, __HIP_MEMORY_SCOPE_AGENT);
}

// ---------------------------------------------------------------------------
// Degree: deg[c*N+n] = 1 (self loop) + sum_{e: dst==n} w[e];  then dis = rsqrt
// ---------------------------------------------------------------------------
__global__ void k_deg_init(float* deg) {
    int i = blockIdx.x * blockDim.x + threadIdx.x;
    if (i < C_REL * N_NODES) deg[i] = 1.0f;           // self-loop weight
}

__global__ void k_deg_acc(const int* __restrict__ ei, const float* __restrict__ ew,
                          float* __restrict__ deg) {
    int i = blockIdx.x * blockDim.x + threadIdx.x;
    if (i >= C_REL * E_EDGES) return;
    int c = i / E_EDGES;
    int e = i - c * E_EDGES;
    int dst = ei[(size_t)c * 2 * E_EDGES + E_EDGES + e];
    atomic_add_f32(&deg[c * N_NODES + dst], ew[(size_t)c * E_EDGES + e]);
}

__global__ void k_rsqrt(float* deg) {
    int i = blockIdx.x * blockDim.x + threadIdx.x;
    if (i < C_REL * N_NODES) deg[i] = rsqrtf(deg[i]); // deg >= 1 always (self loops)
}

// ---------------------------------------------------------------------------
// WMMA fp32 GEMM:  Out[c*N + n, j] = sum_k A[c*N + n, k] * B[c, k, j]
// One wave computes one 16x16 output tile, K stepped by 4 with
// V_WMMA_F32_16X16X4_F32 (exact fp32).
// A 16x4 layout: VGPR0 = K0 (lanes0-15) / K2 (lanes16-31); VGPR1 = K1 / K3
// B 4x16 layout: row-striped across lanes, same half-wave K split
// D 16x16 layout: VGPR i -> row (i + 8*half), col (lane & 15)
// ---------------------------------------------------------------------------
__global__ void k_gemm_wmma(const float* __restrict__ A, const float* __restrict__ B,
                            float* __restrict__ Out, int K, int J) {
    const int tilesN = N_NODES / 16;
    const int tilesJ = J / 16;
    const int total  = C_REL * tilesN * tilesJ;
    int wave = (blockIdx.x * blockDim.x + threadIdx.x) >> 5;
    if (wave >= total) return;                         // whole waves exit: EXEC all-1s

    int c  = wave / (tilesN * tilesJ);
    int r  = wave - c * (tilesN * tilesJ);
    int tn = r / tilesJ;
    int tj = r - tn * tilesJ;

    int lane = threadIdx.x & 31;
    int hx   = lane >> 4;                              // half-wave: 0 or 1
    int m    = lane & 15;

    const size_t arow = ((size_t)c * N_NODES + tn * 16 + m) * (size_t)K;
    const float* Bc   = B + (size_t)c * K * J + tj * 16 + m;

    v8f acc = {};
    for (int k = 0; k < K; k += 4) {
        v2f a = *(const v2f*)(A + arow + k + 2 * hx);  // contiguous, 8B aligned
        v2f b;
        b.x = Bc[(size_t)(k + 2 * hx)     * J];
        b.y = Bc[(size_t)(k + 2 * hx + 1) * J];
        acc = __builtin_amdgcn_wmma_f32_16x16x4_f32(
            /*neg_a=*/false, a, /*neg_b=*/false, b,
            /*c_mod=*/(short)0, acc, /*reuse_a=*/false, /*reuse_b=*/false);
    }

    float* Oc = Out + ((size_t)c * N_NODES + tn * 16 + 8 * hx) * (size_t)J + tj * 16 + m;
#pragma unroll
    for (int i = 0; i < 8; ++i) Oc[(size_t)i * J] = acc[i];
}

// ---------------------------------------------------------------------------
// Layer-1 aggregation
// ---------------------------------------------------------------------------
// out[n,:] = dis[n]^2 * h[n,:]   (self-loop contribution, also initializes buffer)
__global__ void k_selfloop_128(const float* __restrict__ h, const float* __restrict__ dis,
                               float* __restrict__ out) {
    size_t i = (size_t)blockIdx.x * blockDim.x + threadIdx.x;
    if (i >= (size_t)C_REL * N_NODES * 128) return;
    size_t node = i >> 7;
    float d = dis[node];
    out[i] = d * d * h[i];
}

// wave per edge; lane covers 4 features (float4 gather, 4 f32 atomics)
__global__ void k_agg_edges_128(const int* __restrict__ ei, const float* __restrict__ ew,
                                const float* __restrict__ dis,
                                const float* __restrict__ h, float* __restrict__ out) {
    int wave = (blockIdx.x * blockDim.x + threadIdx.x) >> 5;
    if (wave >= C_REL * E_EDGES) return;
    int c = wave / E_EDGES;
    int e = wave - c * E_EDGES;
    int lane = threadIdx.x & 31;

    int src = ei[(size_t)c * 2 * E_EDGES + e];
    int dst = ei[(size_t)c * 2 * E_EDGES + E_EDGES + e];
    float norm = dis[c * N_NODES + src] * ew[(size_t)c * E_EDGES + e] * dis[c * N_NODES + dst];

    const float4 hv = *(const float4*)(h + ((size_t)c * N_NODES + src) * 128 + lane * 4);
    float* o = out + ((size_t)c * N_NODES + dst) * 128 + lane * 4;
    atomic_add_f32(o + 0, norm * hv.x);
    atomic_add_f32(o + 1, norm * hv.y);
    atomic_add_f32(o + 2, norm * hv.z);
    atomic_add_f32(o + 3, norm * hv.w);
}

// a1 = relu(a1 + b1)   in place
__global__ void k_bias_relu_128(float* __restrict__ a, const float* __restrict__ b1) {
    size_t i = (size_t)blockIdx.x * blockDim.x + threadIdx.x;
    if (i >= (size_t)C_REL * N_NODES * 128) return;
    int j = (int)(i & 127);
    int c = (int)(i / ((size_t)N_NODES * 128));
    float v = a[i] + b1[c * H_F + j];
    a[i] = v > 0.0f ? v : 0.0f;
}

// ---------------------------------------------------------------------------
// Layer-2 aggregation (64 features)
// ---------------------------------------------------------------------------
// out[n,:] = dis[n]^2 * h2[n,:] + b2  (self-loop + bias; fully initializes d_out)
__global__ void k_selfloop_bias_64(const float* __restrict__ h2, const float* __restrict__ dis,
                                   const float* __restrict__ b2, float* __restrict__ out) {
    size_t i = (size_t)blockIdx.x * blockDim.x + threadIdx.x;
    if (i >= (size_t)C_REL * N_NODES * 64) return;
    size_t node = i >> 6;
    int j = (int)(i & 63);
    int c = (int)(i / ((size_t)N_NODES * 64));
    float d = dis[node];
    out[i] = d * d * h2[i] + b2[c * DOUT_F + j];
}

// wave per edge; lane covers 2 features
__global__ void k_agg_edges_64(const int* __restrict__ ei, const float* __restrict__ ew,
                               const float* __restrict__ dis,
                               const float* __restrict__ h, float* __restrict__ out) {
    int wave = (blockIdx.x * blockDim.x + threadIdx.x) >> 5;
    if (wave >= C_REL * E_EDGES) return;
    int c = wave / E_EDGES;
    int e = wave - c * E_EDGES;
    int lane = threadIdx.x & 31;

    int src = ei[(size_t)c * 2 * E_EDGES + e];
    int dst = ei[(size_t)c * 2 * E_EDGES + E_EDGES + e];
    float norm = dis[c * N_NODES + src] * ew[(size_t)c * E_EDGES + e] * dis[c * N_NODES + dst];

    const float2 hv = *(const float2*)(h + ((size_t)c * N_NODES + src) * 64 + lane * 2);
    float* o = out + ((size_t)c * N_NODES + dst) * 64 + lane * 2;
    atomic_add_f32(o + 0, norm * hv.x);
    atomic_add_f32(o + 1, norm * hv.y);
}

// ---------------------------------------------------------------------------
extern "C" void kernel_launch(void* const* d_in, const int* in_sizes, int n_in,
                              void* d_out, int out_size, void* d_ws, size_t ws_size,
                              hipStream_t stream) {
    const float* x  = (const float*)d_in[0];   // (C,N,DIN)
    const int*   ei = (const int*)  d_in[1];   // (C,2,E)
    const float* ew = (const float*)d_in[2];   // (C,E)
    const float* W1 = (const float*)d_in[3];   // (C,DIN,H)
    const float* b1 = (const float*)d_in[4];   // (C,H)
    const float* W2 = (const float*)d_in[5];   // (C,H,DOUT)
    const float* b2 = (const float*)d_in[6];   // (C,DOUT)
    float* out = (float*)d_out;                // (C,N,DOUT)

    // Workspace layout (~257 MB):
    //   [0, 1MB)        dis (C*N f32, holds deg then deg^-1/2)
    //   [1MB, +128MB)   h1  (C*N*128) — reused as h2 (C*N*64) after layer 1
    //   [+128MB,+128MB) a1  (C*N*128) — layer-1 aggregation, relu'd in place
    char*  ws  = (char*)d_ws;
    float* dis = (float*)ws;
    float* h1  = (float*)(ws + (1u << 20));
    float* a1  = h1 + (size_t)C_REL * N_NODES * H_F;
    float* h2  = h1;                           // reuse: h1 dead after aggregation

    const int B = 256;
    const int nCN      = C_REL * N_NODES;                 // 250,000
    const int nCE      = C_REL * E_EDGES;                 // 4,000,000
    const long nCN128  = (long)nCN * 128;                 // 32,000,000
    const long nCN64   = (long)nCN * 64;                  // 16,000,000
    const int edgeWaveBlocks = (int)(((long)nCE * 32 + B - 1) / B);   // 500,000

    // degree -> dis
    k_deg_init<<<(nCN + B - 1) / B, B, 0, stream>>>(dis);
    k_deg_acc <<<(nCE + B - 1) / B, B, 0, stream>>>(ei, ew, dis);
    k_rsqrt   <<<(nCN + B - 1) / B, B, 0, stream>>>(dis);

    // layer 1: h1 = x @ W1 ; aggregate ; +b1, relu (in place)
    {
        int waves  = C_REL * (N_NODES / 16) * (H_F / 16);     // 125,000
        int blocks = (int)(((long)waves * 32 + B - 1) / B);   // 15,625
        k_gemm_wmma<<<blocks, B, 0, stream>>>(x, W1, h1, DIN_F, H_F);
    }
    k_selfloop_128 <<<(int)((nCN128 + B - 1) / B), B, 0, stream>>>(h1, dis, a1);
    k_agg_edges_128<<<edgeWaveBlocks, B, 0, stream>>>(ei, ew, dis, h1, a1);
    k_bias_relu_128<<<(int)((nCN128 + B - 1) / B), B, 0, stream>>>(a1, b1);

    // layer 2: h2 = a1 @ W2 ; aggregate into d_out ; +b2 folded into init
    {
        int waves  = C_REL * (N_NODES / 16) * (DOUT_F / 16);  // 62,500
        int blocks = (int)(((long)waves * 32 + B - 1) / B);
        k_gemm_wmma<<<blocks, B, 0, stream>>>(a1, W2, h2, H_F, DOUT_F);
    }
    k_selfloop_bias_64<<<(int)((nCN64 + B - 1) / B), B, 0, stream>>>(h2, dis, b2, out);
    k_agg_edges_64    <<<edgeWaveBlocks, B, 0, stream>>>(ei, ew, dis, h2, out);
}